// SimpMamba_4587025072802
// MI455X (gfx1250) — compile-verified
//
#include <hip/hip_runtime.h>
#include <cmath>

// ---------------- problem constants (from reference) ----------------
#define DMODEL 128
#define DSTATE 256
#define DINNER 256
#define BB     4
#define LLEN   100
#define NROWS  (BB * LLEN)          // 400 token rows

typedef __attribute__((ext_vector_type(2))) float v2f;
typedef __attribute__((ext_vector_type(8))) float v8f;

__device__ __forceinline__ float silu_f(float x) {
    return x / (1.0f + __expf(-x));
}
__device__ __forceinline__ float softplus_f(float x) {
    return (x > 20.0f) ? x : log1pf(__expf(x));
}

// --------------------------------------------------------------------
// Generic Y = X @ W^T (+ bias, optional softplus), fp32 via
// V_WMMA_F32_16X16X4_F32. One wave per 16x16 output tile.
//   X: M x K row-major     W: N x K row-major (torch Linear layout)
//   Y: M x ldy (writes N columns)
// EPI: 0 = (+bias if non-null), 1 = softplus(x + bias)
// --------------------------------------------------------------------
template <int EPI>
__global__ void wmma_gemm_xwt(const float* __restrict__ X,
                              const float* __restrict__ W,
                              const float* __restrict__ bias,
                              float* __restrict__ Y,
                              int M, int N, int K, int ldy)
{
    const int wavesPerBlock = blockDim.x >> 5;
    const int wave = blockIdx.x * wavesPerBlock + (threadIdx.x >> 5);
    const int lane = threadIdx.x & 31;
    const int ntilesN = N >> 4;
    const int totalTiles = (M >> 4) * ntilesN;
    if (wave >= totalTiles) return;

    const int tm = wave / ntilesN;
    const int tn = wave % ntilesN;

    const int half = lane >> 4;   // 0: K pair {0,1}, 1: K pair {2,3}
    const int l15  = lane & 15;

    const float* Arow = X + (size_t)(tm * 16 + l15) * K;  // A row M = l15
    const float* Wrow = W + (size_t)(tn * 16 + l15) * K;  // B col N = l15 (from W^T)

    v8f c = {0.f, 0.f, 0.f, 0.f, 0.f, 0.f, 0.f, 0.f};

    for (int k = 0; k < K; k += 4) {
        // 16x4 A tile: lanes 0-15 -> K=k,k+1 ; lanes 16-31 -> K=k+2,k+3
        v2f a, b;
        a.x = Arow[k + half * 2 + 0];
        a.y = Arow[k + half * 2 + 1];
        // 4x16 B tile, B[kk][n] = W[n][kk], same lane-half K split
        b.x = Wrow[k + half * 2 + 0];
        b.y = Wrow[k + half * 2 + 1];
        c = __builtin_amdgcn_wmma_f32_16x16x4_f32(
                /*neg_a=*/false, a, /*neg_b=*/false, b,
                /*c_mod=*/(short)0, c, /*reuse_a=*/false, /*reuse_b=*/false);
    }

    // C/D layout: VGPR i -> row M = i + half*8, col N = l15
    const int col  = tn * 16 + l15;
    const int row0 = tm * 16 + half * 8;
    const float bv = (bias != nullptr) ? bias[col] : 0.0f;
#pragma unroll
    for (int i = 0; i < 8; ++i) {
        float v = c[i] + bv;
        if (EPI == 1) v = softplus_f(v);
        Y[(size_t)(row0 + i) * ldy + col] = v;
    }
}

// --------------------------------------------------------------------
// Causal depthwise conv (d_conv=4) + SiLU.  x = xz[:, :DINNER].
// One thread per (b, l, d).
// --------------------------------------------------------------------
__global__ void conv_silu_kernel(const float* __restrict__ xz,
                                 const float* __restrict__ conv_w,
                                 const float* __restrict__ conv_b,
                                 float* __restrict__ xc)
{
    int idx = blockIdx.x * blockDim.x + threadIdx.x;
    if (idx >= NROWS * DINNER) return;
    const int d  = idx & (DINNER - 1);
    const int bl = idx >> 8;
    const int l  = bl % LLEN;
    const int b  = bl / LLEN;

    const float* w = conv_w + d * 4;
    float acc = conv_b[d];
#pragma unroll
    for (int j = 0; j < 4; ++j) {
        const int ls = l - 3 + j;
        if (ls >= 0)
            acc += w[j] * xz[(size_t)(b * LLEN + ls) * (2 * DINNER) + d];
    }
    xc[(size_t)bl * DINNER + d] = silu_f(acc);
}

// --------------------------------------------------------------------
// Fused: actual_state = state*exp(dt*A) + dt*B   (A = -exp(A_log))
//        y = einsum("dn,dn->d", actual_state, C[l]) + D1*x ; y *= silu(z)
// One block per (b,l). 8 waves; each wave streams one d-row with float4.
// actual_state is produced AND consumed in registers (single HBM pass).
// --------------------------------------------------------------------
__global__ void fused_state_kernel(const float* __restrict__ state,
                                   const float* __restrict__ A_log,
                                   const float* __restrict__ Bm,
                                   const float* __restrict__ dt,
                                   const float* __restrict__ C_param,
                                   const float* __restrict__ D1,
                                   const float* __restrict__ xc,
                                   const float* __restrict__ xz,
                                   float* __restrict__ actual_state,
                                   float* __restrict__ y)
{
    const int bl   = blockIdx.x;          // b*LLEN + l
    const int l    = bl % LLEN;
    const int wave = threadIdx.x >> 5;
    const int lane = threadIdx.x & 31;

    const float* srow = state        + (size_t)bl * (DINNER * DSTATE);
    float*       orow = actual_state + (size_t)bl * (DINNER * DSTATE);
    const float* Crow = C_param      + (size_t)l  * (DINNER * DSTATE);
    const float4* b4  = (const float4*)(Bm + (size_t)bl * DSTATE);

    for (int d = wave; d < DINNER; d += 8) {
        const float dtv = dt[(size_t)bl * DINNER + d];
        const float4* s4 = (const float4*)(srow  + (size_t)d * DSTATE);
        const float4* a4 = (const float4*)(A_log + (size_t)d * DSTATE);
        const float4* c4 = (const float4*)(Crow  + (size_t)d * DSTATE);
        float4*       o4 = (float4*)(orow + (size_t)d * DSTATE);

        float acc = 0.0f;
#pragma unroll
        for (int ch = 0; ch < 2; ++ch) {
            const int i = ch * 32 + lane;           // float4 index; 64 per row
            const float4 sv = s4[i];
            const float4 av = a4[i];
            const float4 cv = c4[i];
            const float4 bv = b4[i];
            float4 ov;
            ov.x = sv.x * __expf(-dtv * __expf(av.x)) + dtv * bv.x;
            ov.y = sv.y * __expf(-dtv * __expf(av.y)) + dtv * bv.y;
            ov.z = sv.z * __expf(-dtv * __expf(av.z)) + dtv * bv.z;
            ov.w = sv.w * __expf(-dtv * __expf(av.w)) + dtv * bv.w;
            o4[i] = ov;
            acc += ov.x * cv.x + ov.y * cv.y + ov.z * cv.z + ov.w * cv.w;
        }
        // 32-lane reduction
        for (int off = 16; off > 0; off >>= 1)
            acc += __shfl_down(acc, off);

        if (lane == 0) {
            float yy = acc + D1[d] * xc[(size_t)bl * DINNER + d];
            const float zv = xz[(size_t)bl * (2 * DINNER) + DINNER + d];
            yy *= silu_f(zv);
            y[(size_t)bl * DINNER + d] = yy;
        }
    }
}

// --------------------------------------------------------------------
extern "C" void kernel_launch(void* const* d_in, const int* in_sizes, int n_in,
                              void* d_out, int out_size, void* d_ws, size_t ws_size,
                              hipStream_t stream)
{
    (void)in_sizes; (void)n_in; (void)out_size; (void)ws_size;

    const float* hs         = (const float*)d_in[0];   // (4,100,128)
    const float* state      = (const float*)d_in[1];   // (4,100,256,256)
    const float* in_proj_w  = (const float*)d_in[2];   // (512,128)
    const float* in_proj_b  = (const float*)d_in[3];   // (512,)
    const float* conv_w     = (const float*)d_in[4];   // (256,1,4)
    const float* conv_b     = (const float*)d_in[5];   // (256,)
    const float* x_proj_w   = (const float*)d_in[6];   // (512,256)  rows 0..255 -> B
    const float* dt_proj_w  = (const float*)d_in[7];   // (256,256)
    const float* dt_proj_b  = (const float*)d_in[8];   // (256,)
    const float* A_log      = (const float*)d_in[9];   // (256,256)
    const float* D1         = (const float*)d_in[10];  // (256,)
    const float* C_param    = (const float*)d_in[11];  // (100,256,256)
    const float* out_proj_w = (const float*)d_in[12];  // (128,256)
    const float* out_proj_b = (const float*)d_in[13];  // (128,)

    float* out          = (float*)d_out;                       // (400,128)
    float* actual_state = out + (size_t)NROWS * DMODEL;        // (400,256,256)

    float* ws  = (float*)d_ws;
    float* xz  = ws;                                  // 400*512
    float* xc  = xz  + (size_t)NROWS * 512;           // 400*256
    float* Bm  = xc  + (size_t)NROWS * DINNER;        // 400*256
    float* dtb = Bm  + (size_t)NROWS * DSTATE;        // 400*256
    float* yb  = dtb + (size_t)NROWS * DINNER;        // 400*256

    auto gemm_blocks = [](int M, int N) {
        int waves = (M >> 4) * (N >> 4);
        return (waves + 7) / 8;                       // 8 waves / 256-thread block
    };

    // 1) xz = hidden @ in_proj_w^T + b       (400 x 512, K=128)
    wmma_gemm_xwt<0><<<gemm_blocks(NROWS, 2 * DINNER), 256, 0, stream>>>(
        hs, in_proj_w, in_proj_b, xz, NROWS, 2 * DINNER, DMODEL, 2 * DINNER);

    // 2) causal depthwise conv + silu -> xc  (400 x 256)
    conv_silu_kernel<<<(NROWS * DINNER) / 256, 256, 0, stream>>>(
        xz, conv_w, conv_b, xc);

    // 3a) Bm = xc @ x_proj_w[:256]^T         (400 x 256, K=256)
    wmma_gemm_xwt<0><<<gemm_blocks(NROWS, DSTATE), 256, 0, stream>>>(
        xc, x_proj_w, nullptr, Bm, NROWS, DSTATE, DINNER, DSTATE);

    // 3b) dt = softplus(xc @ dt_proj_w^T + b)
    wmma_gemm_xwt<1><<<gemm_blocks(NROWS, DINNER), 256, 0, stream>>>(
        xc, dt_proj_w, dt_proj_b, dtb, NROWS, DINNER, DINNER, DINNER);

    // 4) fused state update + einsum + gating (bandwidth-dominant pass)
    fused_state_kernel<<<NROWS, 256, 0, stream>>>(
        state, A_log, Bm, dtb, C_param, D1, xc, xz, actual_state, yb);

    // 5) out = yb @ out_proj_w^T + b          (400 x 128, K=256)
    wmma_gemm_xwt<0><<<gemm_blocks(NROWS, DMODEL), 256, 0, stream>>>(
        yb, out_proj_w, out_proj_b, out, NROWS, DMODEL, DINNER, DMODEL);
}